// KVMemoryGraft_88510686036001
// MI455X (gfx1250) — compile-verified
//
#include <hip/hip_runtime.h>
#include <hip/hip_bf16.h>
#include <stdint.h>

#define D_MODEL 2048
#define N_ITEMS 8192
#define BB 8
#define SS 4096
#define TOPK 64
#define TEMPERATURE 0.03f
#define STRENGTH 16.0f
#define THRESHOLD 0.85f
#define GATE_SHARPNESS 40.0f

typedef float v2f __attribute__((ext_vector_type(2)));
typedef float v8f __attribute__((ext_vector_type(8)));

// ---- workspace byte offsets ----
#define OFF_Q       0          // 8*2048 f32 = 65536 B  (query accum / mean)
#define OFF_QINV    65536      // 8 f32
#define OFF_LENF    65792      // 8 f32
#define OFF_LAST    66048      // 8 int
#define OFF_SIMS    66560      // 8*8192 f32 = 262144 B
#define OFF_SELIDX  328704     // 8*64 int = 2048 B
#define OFF_SELW    330752     // 8*64 f32 = 2048 B
#define OFF_SELCNT  332800     // 8 int
#define OFF_GATEC   333056     // 8 f32
// total ~333 KB

// ---------------------------------------------------------------------------
// K0: per-batch mask length + last index; zero query accumulators
// ---------------------------------------------------------------------------
__global__ void __launch_bounds__(256)
k_init(const unsigned char* __restrict__ mask, float* __restrict__ q,
       float* __restrict__ lenf, int* __restrict__ last) {
  int b = blockIdx.x, t = threadIdx.x;
  __shared__ int sh[256];
  int cnt = 0;
  for (int s = t; s < SS; s += 256) cnt += mask[b * SS + s] ? 1 : 0;
  sh[t] = cnt; __syncthreads();
  for (int o = 128; o > 0; o >>= 1) { if (t < o) sh[t] += sh[t + o]; __syncthreads(); }
  if (t == 0) {
    int c = sh[0];
    lenf[b] = fmaxf((float)c, 1.0f);
    last[b] = (c > 0 ? c : 1) - 1;
  }
  for (int d = t; d < D_MODEL; d += 256) q[b * D_MODEL + d] = 0.0f;
}

// ---------------------------------------------------------------------------
// K1: fused x -> out copy + masked column-sum accumulation (x read ONCE)
// grid = (SS/64, BB), 256 threads. Each thread owns 8 disjoint columns.
// ---------------------------------------------------------------------------
__global__ void __launch_bounds__(256)
k_copy_qaccum(const float4* __restrict__ x4, const unsigned char* __restrict__ mask,
              float4* __restrict__ out4, float* __restrict__ q) {
  const int D4 = D_MODEL / 4;  // 512
  int b = blockIdx.y;
  int s0 = blockIdx.x * 64;
  int t = threadIdx.x;
  float4 a0 = {0.f, 0.f, 0.f, 0.f}, a1 = {0.f, 0.f, 0.f, 0.f};
  size_t base = ((size_t)b * SS + s0) * D4;
  for (int s = 0; s < 64; ++s) {
    float m = mask[b * SS + s0 + s] ? 1.0f : 0.0f;
    size_t row = base + (size_t)s * D4;
    float4 v0 = x4[row + t];
    float4 v1 = x4[row + t + 256];
    out4[row + t] = v0;
    out4[row + t + 256] = v1;
    a0.x += m * v0.x; a0.y += m * v0.y; a0.z += m * v0.z; a0.w += m * v0.w;
    a1.x += m * v1.x; a1.y += m * v1.y; a1.z += m * v1.z; a1.w += m * v1.w;
  }
  float* qb = q + b * D_MODEL;
  atomicAdd(&qb[4 * t + 0], a0.x);
  atomicAdd(&qb[4 * t + 1], a0.y);
  atomicAdd(&qb[4 * t + 2], a0.z);
  atomicAdd(&qb[4 * t + 3], a0.w);
  atomicAdd(&qb[4 * (t + 256) + 0], a1.x);
  atomicAdd(&qb[4 * (t + 256) + 1], a1.y);
  atomicAdd(&qb[4 * (t + 256) + 2], a1.z);
  atomicAdd(&qb[4 * (t + 256) + 3], a1.w);
}

// ---------------------------------------------------------------------------
// K2: finalize queries: divide by length, compute 1/max(||q||, eps)
// ---------------------------------------------------------------------------
__global__ void __launch_bounds__(256)
k_qfinal(float* __restrict__ q, const float* __restrict__ lenf, float* __restrict__ qinv) {
  int b = blockIdx.x, t = threadIdx.x;
  float inv_len = 1.0f / lenf[b];
  float ss = 0.0f;
  for (int d = t; d < D_MODEL; d += 256) {
    float v = q[b * D_MODEL + d] * inv_len;
    q[b * D_MODEL + d] = v;
    ss += v * v;
  }
  __shared__ float sh[256];
  sh[t] = ss; __syncthreads();
  for (int o = 128; o > 0; o >>= 1) { if (t < o) sh[t] += sh[t + o]; __syncthreads(); }
  if (t == 0) qinv[b] = 1.0f / fmaxf(sqrtf(sh[0]), 1e-12f);
}

// ---------------------------------------------------------------------------
// K3: sims = normalize(q) @ normalize(keys)^T via V_WMMA_F32_16X16X4_F32.
// grid = N_ITEMS/128 blocks of 256 threads (8 waves); each wave owns a
// 16-column tile. Keys streamed through LDS (stride 65 = conflict-free);
// per-key ||k||^2 accumulated on the fly (keys read once, fused norm).
// ---------------------------------------------------------------------------
__global__ void __launch_bounds__(256)
k_sims_wmma(const float* __restrict__ keys, const float* __restrict__ q,
            const float* __restrict__ qinv, float* __restrict__ sims) {
  __shared__ float qs[16 * 65];
  __shared__ float ks[128 * 65];
  __shared__ float ssh[256];
  __shared__ float kinv_sh[128];

  int t = threadIdx.x;
  int wave = t >> 5, lane = t & 31;
  int hl = lane >> 4, ll = lane & 15;
  int n_base = blockIdx.x * 128;

  v8f c = {};
  float ssacc = 0.0f;
  int ssrow = t >> 1, sspart = t & 1;

  for (int kc = 0; kc < D_MODEL; kc += 64) {
    for (int i = t; i < 16 * 64; i += 256) {
      int r = i >> 6, cc = i & 63;
      qs[r * 65 + cc] = (r < BB) ? q[r * D_MODEL + kc + cc] : 0.0f;
    }
    for (int i = t; i < 128 * 64; i += 256) {
      int r = i >> 6, cc = i & 63;
      ks[r * 65 + cc] = keys[(size_t)(n_base + r) * D_MODEL + kc + cc];
    }
    __syncthreads();

    // fused key-norm partial sums (2 threads per key row)
    {
      const float* kr = &ks[ssrow * 65 + sspart * 32];
      float s2 = 0.0f;
      #pragma unroll
      for (int cc = 0; cc < 32; ++cc) { float v = kr[cc]; s2 += v * v; }
      ssacc += s2;
    }

    #pragma unroll
    for (int kk = 0; kk < 64; kk += 4) {
      v2f a, bmat;
      // A (16x4 f32): VGPR0 = K{0|2}, VGPR1 = K{1|3} per half-wave
      a.x = qs[ll * 65 + kk + hl * 2];
      a.y = qs[ll * 65 + kk + hl * 2 + 1];
      // B (4x16 f32): row (K) striped across lanes within a VGPR
      bmat.x = ks[(wave * 16 + ll) * 65 + kk + hl * 2];
      bmat.y = ks[(wave * 16 + ll) * 65 + kk + hl * 2 + 1];
      c = __builtin_amdgcn_wmma_f32_16x16x4_f32(
          false, a, false, bmat, (short)0, c, false, false);
    }
    __syncthreads();
  }

  ssh[t] = ssacc;
  __syncthreads();
  if (t < 128)
    kinv_sh[t] = 1.0f / fmaxf(sqrtf(ssh[2 * t] + ssh[2 * t + 1]), 1e-12f);
  __syncthreads();

  // C/D layout: VGPR j holds M = j + 8*half, N = lane%16. Only M<8 is real.
  if (hl == 0) {
    int ncol = wave * 16 + ll;
    int n = n_base + ncol;
    float sk = kinv_sh[ncol];
    #pragma unroll
    for (int j = 0; j < 8; ++j)
      sims[j * N_ITEMS + n] = c[j] * qinv[j] * sk;
  }
}

// ---------------------------------------------------------------------------
// K4: per-row top-64 via bisection on count, softmax weights, gate coeff.
// One block per batch row; sims row lives in LDS (32 KB).
// ---------------------------------------------------------------------------
__global__ void __launch_bounds__(256)
k_topk(const float* __restrict__ sims, int* __restrict__ sel_idx,
       float* __restrict__ sel_w, int* __restrict__ sel_cnt,
       float* __restrict__ gatec) {
  int b = blockIdx.x, t = threadIdx.x;
  __shared__ float srow[N_ITEMS];
  __shared__ float shf[256];
  __shared__ int shcnt;
  __shared__ float shZ;

  for (int i = t; i < N_ITEMS; i += 256) srow[i] = sims[b * N_ITEMS + i];
  __syncthreads();

  // max reduction (sims are cosine sims, in [-1, 1])
  float mx = -2.0f;
  for (int i = t; i < N_ITEMS; i += 256) mx = fmaxf(mx, srow[i]);
  shf[t] = mx; __syncthreads();
  for (int o = 128; o > 0; o >>= 1) { if (t < o) shf[t] = fmaxf(shf[t], shf[t + o]); __syncthreads(); }
  float maxv = shf[0];
  __syncthreads();

  // bisection: maintain count(s > lo) >= TOPK, count(s > hi) < TOPK
  float lo = -2.0f, hi = maxv;
  for (int it = 0; it < 48; ++it) {
    float mid = 0.5f * (lo + hi);
    if (t == 0) shcnt = 0;
    __syncthreads();
    int c = 0;
    for (int i = t; i < N_ITEMS; i += 256) c += (srow[i] > mid) ? 1 : 0;
    atomicAdd(&shcnt, c);
    __syncthreads();
    int cnt = shcnt;
    __syncthreads();
    if (cnt >= TOPK) lo = mid; else hi = mid;
  }

  // compact selected (s > lo), capped at TOPK; accumulate softmax partition Z
  if (t == 0) { shcnt = 0; shZ = 0.0f; }
  __syncthreads();
  const float invT = 1.0f / TEMPERATURE;
  for (int i = t; i < N_ITEMS; i += 256) {
    float s = srow[i];
    if (s > lo) {
      int pos = atomicAdd(&shcnt, 1);
      if (pos < TOPK) {
        float w = __expf((s - maxv) * invT);
        sel_idx[b * TOPK + pos] = i;
        sel_w[b * TOPK + pos] = w;
        atomicAdd(&shZ, w);
      }
    }
  }
  __syncthreads();
  if (t == 0) {
    int c = shcnt; if (c > TOPK) c = TOPK;
    sel_cnt[b] = c;
    float gate = STRENGTH / (1.0f + __expf(-(maxv - THRESHOLD) * GATE_SHARPNESS));
    gatec[b] = gate / fmaxf(shZ, 1e-30f);
  }
}

// ---------------------------------------------------------------------------
// K5: out[b, last_b, :] += gatec[b] * sum_i w_i * values[idx_i, :]
// grid = (D/256, B); values reads coalesced over d.
// ---------------------------------------------------------------------------
__global__ void __launch_bounds__(256)
k_apply(const float* __restrict__ values, const int* __restrict__ sel_idx,
        const float* __restrict__ sel_w, const int* __restrict__ sel_cnt,
        const float* __restrict__ gatec, const int* __restrict__ last,
        float* __restrict__ out) {
  int b = blockIdx.y;
  int d = blockIdx.x * 256 + threadIdx.x;
  int c = sel_cnt[b];
  float acc = 0.0f;
  for (int i = 0; i < c; ++i) {
    acc += sel_w[b * TOPK + i] *
           values[(size_t)sel_idx[b * TOPK + i] * D_MODEL + d];
  }
  size_t o = ((size_t)b * SS + (size_t)last[b]) * D_MODEL + d;
  out[o] += gatec[b] * acc;
}

// ---------------------------------------------------------------------------
extern "C" void kernel_launch(void* const* d_in, const int* in_sizes, int n_in,
                              void* d_out, int out_size, void* d_ws, size_t ws_size,
                              hipStream_t stream) {
  const float*         x      = (const float*)d_in[0];
  const unsigned char* mask   = (const unsigned char*)d_in[1];  // jnp bool = 1 byte
  const float*         keys   = (const float*)d_in[2];
  const float*         values = (const float*)d_in[3];
  float*               out    = (float*)d_out;

  char* ws = (char*)d_ws;
  float* q       = (float*)(ws + OFF_Q);
  float* qinv    = (float*)(ws + OFF_QINV);
  float* lenf    = (float*)(ws + OFF_LENF);
  int*   last    = (int*)  (ws + OFF_LAST);
  float* sims    = (float*)(ws + OFF_SIMS);
  int*   sel_idx = (int*)  (ws + OFF_SELIDX);
  float* sel_w   = (float*)(ws + OFF_SELW);
  int*   sel_cnt = (int*)  (ws + OFF_SELCNT);
  float* gatec   = (float*)(ws + OFF_GATEC);

  k_init<<<BB, 256, 0, stream>>>(mask, q, lenf, last);
  k_copy_qaccum<<<dim3(SS / 64, BB), 256, 0, stream>>>(
      (const float4*)x, mask, (float4*)out, q);
  k_qfinal<<<BB, 256, 0, stream>>>(q, lenf, qinv);
  k_sims_wmma<<<N_ITEMS / 128, 256, 0, stream>>>(keys, q, qinv, sims);
  k_topk<<<BB, 256, 0, stream>>>(sims, sel_idx, sel_w, sel_cnt, gatec);
  k_apply<<<dim3(D_MODEL / 256, BB), 256, 0, stream>>>(
      values, sel_idx, sel_w, sel_cnt, gatec, last, out);
}